// EnhancedUltra_27015344291950
// MI455X (gfx1250) — compile-verified
//
#include <hip/hip_runtime.h>
#include <math.h>

#define R_DIM 128
#define D_DIM 64
#define FEAT_DIM 132   // 2*D + 4

typedef __attribute__((ext_vector_type(2))) float v2f;
typedef __attribute__((ext_vector_type(8))) float v8f;

static __device__ __forceinline__ v8f wmma4(v2f a, v2f b, v8f c) {
  // V_WMMA_F32_16X16X4_F32 : D(16x16,f32) = A(16x4,f32) * B(4x16,f32) + C
  return __builtin_amdgcn_wmma_f32_16x16x4_f32(false, a, false, b, (short)0, c, false, false);
}

// ---------------- workspace init (N only known on device) ----------------
__global__ void k_init(int* __restrict__ map, unsigned* __restrict__ slotHist,
                       unsigned* __restrict__ relCounts,
                       const int* __restrict__ nn_ptr, int B) {
  long N = (long)(*nn_ptr);
  long total = N + (long)B * R_DIM + R_DIM;
  long stride = (long)gridDim.x * blockDim.x;
  for (long i = (long)blockIdx.x * blockDim.x + threadIdx.x; i < total; i += stride) {
    if (i < N)                         map[i] = -1;
    else if (i < N + (long)B * R_DIM)  slotHist[i - N] = 0u;
    else                               relCounts[i - N - (long)B * R_DIM] = 0u;
  }
}

// ---------------- claim slots for queried entities ----------------
__global__ void k_assign(int* __restrict__ map, const int* __restrict__ qe, int B) {
  int b = blockIdx.x * blockDim.x + threadIdx.x;
  if (b < B) atomicCAS(&map[qe[b]], -1, b);   // first b claims; duplicates share a slot
}

// ---------------- single streaming edge pass ----------------
__global__ void k_edges(const int* __restrict__ src, const int* __restrict__ dst,
                        const int* __restrict__ et, const int* __restrict__ map,
                        unsigned* __restrict__ slotHist, unsigned* __restrict__ relCounts,
                        int E) {
  __shared__ unsigned sh[R_DIM];
  for (int i = threadIdx.x; i < R_DIM; i += blockDim.x) sh[i] = 0u;
  __syncthreads();
  long stride = (long)gridDim.x * blockDim.x;
  for (long e = (long)blockIdx.x * blockDim.x + threadIdx.x; e < E; e += stride) {
    int s = src[e], d = dst[e], t = et[e];
    atomicAdd(&sh[t], 1u);                       // rel_counts (all edges)
    int ms = map[s];
    if (ms >= 0) atomicAdd(&slotHist[(long)ms * R_DIM + t], 1u);   // src always
    if (s != d) {
      int md = map[d];
      if (md >= 0) atomicAdd(&slotHist[(long)md * R_DIM + t], 1u); // dst only if not loop
    }
  }
  __syncthreads();
  for (int i = threadIdx.x; i < R_DIM; i += blockDim.x)
    if (sh[i]) atomicAdd(&relCounts[i], sh[i]);
}

// ---------------- per-query feature build (ent_sum GEMV + stats) ----------------
__global__ void k_feat(const float* __restrict__ relemb, const int* __restrict__ qr,
                       const int* __restrict__ qe, const int* __restrict__ map,
                       const unsigned* __restrict__ slotHist,
                       const unsigned* __restrict__ relCounts,
                       const int* __restrict__ nn_ptr,
                       float* __restrict__ feat, float Ef) {
  __shared__ float mq[R_DIM];
  int b = blockIdx.x;
  int t = threadIdx.x;                       // 64 threads, t == output dim d
  int slot = map[qe[b]];
  for (int r = t; r < R_DIM; r += 64)
    mq[r] = (float)slotHist[(long)slot * R_DIM + r];
  __syncthreads();

  float deg = 0.f;
  #pragma unroll 8
  for (int r = 0; r < R_DIM; ++r) deg += mq[r];          // deg_q = row-sum of hist

  const float* eb = relemb + (long)b * R_DIM * D_DIM;
  float acc = 0.f;
  #pragma unroll 4
  for (int r = 0; r < R_DIM; ++r) acc += mq[r] * eb[r * D_DIM + t];

  int q = qr[b];
  float inv = 1.0f / fmaxf(deg, 1.0f);
  float* fb = feat + (long)b * FEAT_DIM;
  fb[t]          = eb[q * D_DIM + t];        // rel_emb
  fb[D_DIM + t]  = acc * inv;                // entity_emb
  if (t == 0) {
    float rfn  = fminf((float)relCounts[q] / Ef, 1.0f);
    float edn  = fminf(deg / Ef, 1.0f);
    float fN   = (float)(*nn_ptr);
    float dens = fminf(Ef / (fN * fN), 1.0f);
    fb[2 * D_DIM + 0] = rfn;
    fb[2 * D_DIM + 1] = edn;
    fb[2 * D_DIM + 2] = rfn;
    fb[2 * D_DIM + 3] = dens;
  }
}

// ---------------- gated MLP: f32 WMMA, one wave per 16-query tile ----------------
#define SH1S 68   // padded LDS strides (avoid 16-way bank conflicts)
#define SH2S 36
#define SGS  20

__global__ __launch_bounds__(32) void k_mlp(
    const float* __restrict__ feat,
    const float* __restrict__ W1,  const float* __restrict__ b1,
    const float* __restrict__ W2,  const float* __restrict__ b2,
    const float* __restrict__ Wg1, const float* __restrict__ bg1,
    const float* __restrict__ Wg2, const float* __restrict__ bg2,
    float* __restrict__ out) {
  __shared__ float sF[16 * FEAT_DIM];
  __shared__ float sH1[16 * SH1S];
  __shared__ float sH2[16 * SH2S];
  __shared__ float sG[16 * SGS];

  int lane = threadIdx.x;
  int m0 = blockIdx.x * 16;
  for (int i = lane; i < 16 * FEAT_DIM; i += 32)
    sF[i] = feat[(long)m0 * FEAT_DIM + i];
  __syncthreads();

  int hl  = lane >> 4;      // half-wave select (K-pair / M+8)
  int nlo = lane & 15;

  // ---- layer 1: [16,132] @ [132,64], K = 33 steps of 4, N = 4 tiles ----
  v8f a0 = {0,0,0,0,0,0,0,0};
  v8f a1 = a0, a2 = a0, a3 = a0;
  for (int kc = 0; kc < 33; ++kc) {
    int k0 = kc * 4 + 2 * hl;
    v2f a; a.x = sF[nlo * FEAT_DIM + k0]; a.y = sF[nlo * FEAT_DIM + k0 + 1];
    const float* w0 = W1 + k0 * 64;
    const float* w1 = W1 + (k0 + 1) * 64;
    v2f bb;
    bb.x = w0[nlo];      bb.y = w1[nlo];      a0 = wmma4(a, bb, a0);
    bb.x = w0[16 + nlo]; bb.y = w1[16 + nlo]; a1 = wmma4(a, bb, a1);
    bb.x = w0[32 + nlo]; bb.y = w1[32 + nlo]; a2 = wmma4(a, bb, a2);
    bb.x = w0[48 + nlo]; bb.y = w1[48 + nlo]; a3 = wmma4(a, bb, a3);
  }
  #pragma unroll
  for (int j = 0; j < 8; ++j) {
    int row = j + 8 * hl;
    sH1[row * SH1S + nlo]      = fmaxf(a0[j] + b1[nlo],      0.f);
    sH1[row * SH1S + 16 + nlo] = fmaxf(a1[j] + b1[16 + nlo], 0.f);
    sH1[row * SH1S + 32 + nlo] = fmaxf(a2[j] + b1[32 + nlo], 0.f);
    sH1[row * SH1S + 48 + nlo] = fmaxf(a3[j] + b1[48 + nlo], 0.f);
  }
  __syncthreads();

  // ---- layer 2: [16,64] @ [64,32] ----
  v8f d0 = {0,0,0,0,0,0,0,0};
  v8f d1 = d0;
  for (int kc = 0; kc < 16; ++kc) {
    int k0 = kc * 4 + 2 * hl;
    v2f a; a.x = sH1[nlo * SH1S + k0]; a.y = sH1[nlo * SH1S + k0 + 1];
    v2f bb;
    bb.x = W2[k0 * 32 + nlo];      bb.y = W2[(k0 + 1) * 32 + nlo];      d0 = wmma4(a, bb, d0);
    bb.x = W2[k0 * 32 + 16 + nlo]; bb.y = W2[(k0 + 1) * 32 + 16 + nlo]; d1 = wmma4(a, bb, d1);
  }
  #pragma unroll
  for (int j = 0; j < 8; ++j) {
    int row = j + 8 * hl;
    sH2[row * SH2S + nlo]      = fmaxf(d0[j] + b2[nlo],      0.f);
    sH2[row * SH2S + 16 + nlo] = fmaxf(d1[j] + b2[16 + nlo], 0.f);
  }
  __syncthreads();

  // ---- layer 3: [16,32] @ [32,16] ----
  v8f g0 = {0,0,0,0,0,0,0,0};
  for (int kc = 0; kc < 8; ++kc) {
    int k0 = kc * 4 + 2 * hl;
    v2f a; a.x = sH2[nlo * SH2S + k0]; a.y = sH2[nlo * SH2S + k0 + 1];
    v2f bb; bb.x = Wg1[k0 * 16 + nlo]; bb.y = Wg1[(k0 + 1) * 16 + nlo];
    g0 = wmma4(a, bb, g0);
  }
  #pragma unroll
  for (int j = 0; j < 8; ++j) {
    int row = j + 8 * hl;
    sG[row * SGS + nlo] = fmaxf(g0[j] + bg1[nlo], 0.f);
  }
  __syncthreads();

  // ---- layer 4: [16,16] @ [16,1] + sigmoid ----
  if (lane < 16) {
    float acc = bg2[0];
    #pragma unroll
    for (int k = 0; k < 16; ++k) acc += sG[lane * SGS + k] * Wg2[k];
    out[m0 + lane] = 1.0f / (1.0f + expf(-acc));
  }
}

extern "C" void kernel_launch(void* const* d_in, const int* in_sizes, int n_in,
                              void* d_out, int out_size, void* d_ws, size_t ws_size,
                              hipStream_t stream) {
  const float* relemb = (const float*)d_in[0];
  const int*   qr     = (const int*)d_in[1];
  const int*   qe     = (const int*)d_in[2];
  const int*   eidx   = (const int*)d_in[3];
  const int*   et     = (const int*)d_in[4];
  const int*   nn     = (const int*)d_in[5];
  const float* W1  = (const float*)d_in[6];
  const float* b1  = (const float*)d_in[7];
  const float* W2  = (const float*)d_in[8];
  const float* b2  = (const float*)d_in[9];
  const float* Wg1 = (const float*)d_in[10];
  const float* bg1 = (const float*)d_in[11];
  const float* Wg2 = (const float*)d_in[12];
  const float* bg2 = (const float*)d_in[13];

  int B = in_sizes[1];          // 1024
  int E = in_sizes[4];          // 6.4M
  const int* src = eidx;
  const int* dst = eidx + E;

  // workspace layout: fixed-size pieces first, N-sized map at the tail
  char* ws = (char*)d_ws;
  size_t off = 0;
  unsigned* slotHist  = (unsigned*)(ws + off); off += (size_t)B * R_DIM * sizeof(unsigned);
  unsigned* relCounts = (unsigned*)(ws + off); off += (size_t)R_DIM * sizeof(unsigned);
  float*    feat      = (float*)   (ws + off); off += (size_t)B * FEAT_DIM * sizeof(float);
  int*      map       = (int*)     (ws + off); // N ints, N read on device

  k_init  <<<2048, 256, 0, stream>>>(map, slotHist, relCounts, nn, B);
  k_assign<<<(B + 255) / 256, 256, 0, stream>>>(map, qe, B);
  k_edges <<<1536, 256, 0, stream>>>(src, dst, et, map, slotHist, relCounts, E);
  k_feat  <<<B, 64, 0, stream>>>(relemb, qr, qe, map, slotHist, relCounts, nn, feat, (float)E);
  k_mlp   <<<B / 16, 32, 0, stream>>>(feat, W1, b1, W2, b2, Wg1, bg1, Wg2, bg2, (float*)d_out);
}